// PVMLayer_14310831031043
// MI455X (gfx1250) — compile-verified
//
#include <hip/hip_runtime.h>
#include <hip/hip_bf16.h>

// ---- problem constants (from reference) ----
#define B_SZ 8
#define C_SZ 128
#define L_SZ 4096          // 64*64
#define NB 4               // branches
#define DM 32              // d_model per branch
#define DI 64              // d_inner
#define DS 16              // d_state
#define OUT_SZ 128
#define NCH 16             // scan chunks
#define CHL 256            // scan chunk length (L/NCH)

typedef __attribute__((ext_vector_type(2))) float v2f;
typedef __attribute__((ext_vector_type(8))) float v8f;

__device__ __forceinline__ float rcp_fast(float x) { return __builtin_amdgcn_rcpf(x); }
__device__ __forceinline__ float siluf(float x) { return x * rcp_fast(1.0f + __expf(-x)); }
__device__ __forceinline__ float softplusf(float x) { return (x > 20.0f) ? x : log1pf(__expf(x)); }
__device__ __forceinline__ float sigmoidf_(float x) { return rcp_fast(1.0f + __expf(-x)); }

// CDNA5 native fp32 WMMA: D(16x16,f32) = A(16x4,f32) x B(4x16,f32) + C
// A layout: lanes 0-15 -> M=lane, K={0,1}; lanes 16-31 -> M=lane-16, K={2,3}
// B layout (mirror): lanes 0-15 -> N=lane, K={0,1}; lanes 16-31 -> N=lane-16, K={2,3}
__device__ __forceinline__ v8f wmma_f32(v2f a, v2f b, v8f c) {
  return __builtin_amdgcn_wmma_f32_16x16x4_f32(false, a, false, b, (short)0, c, false, false);
}

// ---------------- K1: LayerNorm over C, NCHW -> (b,l,c) ----------------
__global__ void k_ln1(const float* __restrict__ x, const float* __restrict__ g,
                      const float* __restrict__ bt, float* __restrict__ xn) {
  int bl = blockIdx.x;                  // b*L + l
  int b = bl >> 12, l = bl & (L_SZ - 1);
  int c = threadIdx.x;                  // 128
  float v = x[(b * C_SZ + c) * L_SZ + l];
  __shared__ float red[C_SZ];
  red[c] = v; __syncthreads();
  for (int o = 64; o > 0; o >>= 1) { if (c < o) red[c] += red[c + o]; __syncthreads(); }
  float mean = red[0] * (1.0f / C_SZ); __syncthreads();
  float d = v - mean;
  red[c] = d * d; __syncthreads();
  for (int o = 64; o > 0; o >>= 1) { if (c < o) red[c] += red[c + o]; __syncthreads(); }
  float var = red[0] * (1.0f / C_SZ);
  xn[bl * C_SZ + c] = d * rsqrtf(var + 1e-5f) * g[c] + bt[c];
}

// -------- K2: in-proj GEMM (fp32 WMMA): per branch (B*L x 32)@(32->128) --------
__global__ void k_inproj(const float* __restrict__ xn, const float* __restrict__ inW,
                         float* __restrict__ xraw, float* __restrict__ zb) {
  __builtin_prefetch(inW, 0, 1);
  int wave = threadIdx.x >> 5, lane = threadIdx.x & 31;
  int gt = blockIdx.x * 8 + wave;       // 4 * 2048 * 8 = 65536 wave-tiles
  int br = gt >> 14;
  int rem = gt & 16383;
  int tr = rem >> 3, tc = rem & 7;
  int hs = lane >> 4, mn = lane & 15;
  const float* arow = xn + (tr * 16 + mn) * C_SZ + br * DM;   // A row m
  const float* brow = inW + (tc * 16 + mn) * DM;              // B col n = W row
  v8f c = {};
  #pragma unroll
  for (int kk = 0; kk < 8; ++kk) {      // K = 32 = 8 x 4
    int kb = kk * 4 + hs * 2;
    v2f a = { arow[kb], arow[kb + 1] };
    v2f b = { brow[kb], brow[kb + 1] };
    c = wmma_f32(a, b, c);
  }
  int base = (br * (B_SZ * L_SZ) + tr * 16) * DI;
  if (tc < 4) {                         // wave-uniform: x half
    #pragma unroll
    for (int r = 0; r < 8; ++r) {
      int m = r + hs * 8;
      xraw[base + m * DI + tc * 16 + mn] = c[r];
    }
  } else {                              // z half -> store silu(z)
    #pragma unroll
    for (int r = 0; r < 8; ++r) {
      int m = r + hs * 8;
      zb[base + m * DI + (tc - 4) * 16 + mn] = siluf(c[r]);
    }
  }
}

// ---- K3: causal depthwise conv(4) + SiLU + x-proj(64->34) + dt-proj fused ----
__global__ void k_conv_proj(const float* __restrict__ xraw, const float* __restrict__ convW,
                            const float* __restrict__ convB, const float* __restrict__ xprojW,
                            const float* __restrict__ dtW, const float* __restrict__ dtB,
                            float* __restrict__ xs, float* __restrict__ dtb,
                            float* __restrict__ Bmb, float* __restrict__ Cmb) {
  int row = blockIdx.x;                 // br*B*L + b*L + l
  int l = row & (L_SZ - 1);
  int d = threadIdx.x;                  // 64
  float acc = convB[d];
  #pragma unroll
  for (int k = 0; k < 4; ++k) {
    int li = l - 3 + k;
    if (li >= 0) acc += convW[d * 4 + k] * xraw[(row - 3 + k) * DI + d];
  }
  float xv = siluf(acc);
  xs[row * DI + d] = xv;
  __shared__ float sx[DI];
  __shared__ float sd[34];
  sx[d] = xv;
  __syncthreads();
  if (d < 34) {
    float s = 0.0f;
    #pragma unroll 8
    for (int dd = 0; dd < DI; ++dd) s += sx[dd] * xprojW[d * DI + dd];
    sd[d] = s;
  }
  __syncthreads();
  float dtpre = sd[0] * dtW[d * 2 + 0] + sd[1] * dtW[d * 2 + 1] + dtB[d];
  dtb[row * DI + d] = softplusf(dtpre);
  if (d < DS) {
    Bmb[row * DS + d] = sd[2 + d];
    Cmb[row * DS + d] = sd[2 + DS + d];
  }
}

// ---- K4a: chunked scan pass 1: per-chunk decay product + partial end state ----
// h_t = exp(dt_t*a)*h_{t-1} + dt_t*B_t*x_t ; decay over chunk = exp(a * sum(dt))
__global__ void __launch_bounds__(1024) k_scan1(
    const float* __restrict__ dtb, const float* __restrict__ xs, const float* __restrict__ Bmb,
    const float* __restrict__ Alog, float* __restrict__ Aprd, float* __restrict__ Hend) {
  int blk = blockIdx.x;                 // flat*NCH + ch
  int flat = blk >> 4, ch = blk & (NCH - 1);
  int d = threadIdx.x >> 4, s = threadIdx.x & 15;
  float a = -__expf(Alog[d * DS + s]);
  int base = flat * L_SZ + ch * CHL;
  float h = 0.0f, dsum = 0.0f;
  for (int t = 0; t < CHL; ++t) {
    int r = base + t;
    float dtv = dtb[r * DI + d];
    float xv  = xs [r * DI + d];
    float Bv  = Bmb[r * DS + s];
    dsum += dtv;
    h = __expf(dtv * a) * h + dtv * Bv * xv;
  }
  int idx = blk * 1024 + threadIdx.x;
  Aprd[idx] = __expf(a * dsum);
  Hend[idx] = h;
}

// ---- K4b: sequential prefix over the 16 chunks -> initial state per chunk ----
__global__ void __launch_bounds__(1024) k_scan2(
    const float* __restrict__ Aprd, const float* __restrict__ Hend, float* __restrict__ Hinit) {
  int flat = blockIdx.x;                // 32
  float hi = 0.0f;
  #pragma unroll
  for (int ch = 0; ch < NCH; ++ch) {
    int idx = (flat * NCH + ch) * 1024 + threadIdx.x;
    Hinit[idx] = hi;
    hi = Aprd[idx] * hi + Hend[idx];
  }
}

// ---- K4c: re-scan each chunk with correct initial state, emit gated y ----
__global__ void __launch_bounds__(1024) k_scan3(
    const float* __restrict__ xs, const float* __restrict__ zb, const float* __restrict__ dtb,
    const float* __restrict__ Bmb, const float* __restrict__ Cmb,
    const float* __restrict__ Alog, const float* __restrict__ Dp,
    const float* __restrict__ Hinit, float* __restrict__ ys) {
  int blk = blockIdx.x;                 // flat*NCH + ch
  int flat = blk >> 4, ch = blk & (NCH - 1);
  int d = threadIdx.x >> 4, s = threadIdx.x & 15;
  float a = -__expf(Alog[d * DS + s]);
  float dpv = Dp[d];
  int base = flat * L_SZ + ch * CHL;
  float h = Hinit[blk * 1024 + threadIdx.x];
  for (int t = 0; t < CHL; ++t) {
    int r = base + t;
    float dtv = dtb[r * DI + d];
    float xv  = xs [r * DI + d];
    float Bv  = Bmb[r * DS + s];
    float Cv  = Cmb[r * DS + s];
    h = __expf(dtv * a) * h + dtv * Bv * xv;
    float p = h * Cv;
    p += __shfl_xor(p, 1, 32);
    p += __shfl_xor(p, 2, 32);
    p += __shfl_xor(p, 4, 32);
    p += __shfl_xor(p, 8, 32);
    if (s == 0) {
      float sz = zb[r * DI + d];        // already silu(z)
      ys[r * DI + d] = (p + xv * dpv) * sz;
    }
  }
}

// -------- K5: out-proj GEMM (fp32 WMMA): (NB*B*L x 64)@(64->32) --------
__global__ void k_outproj(const float* __restrict__ ys, const float* __restrict__ outW,
                          float* __restrict__ M1) {
  __builtin_prefetch(outW, 0, 1);
  int wave = threadIdx.x >> 5, lane = threadIdx.x & 31;
  int gt = blockIdx.x * 8 + wave;       // 8192 row-tiles * 2 col-tiles
  int tr = gt >> 1, tc = gt & 1;
  int hs = lane >> 4, mn = lane & 15;
  const float* arow = ys + (tr * 16 + mn) * DI;
  const float* brow = outW + (tc * 16 + mn) * DI;
  v8f c = {};
  #pragma unroll
  for (int kk = 0; kk < 16; ++kk) {     // K = 64 = 16 x 4
    int kb = kk * 4 + hs * 2;
    v2f a = { arow[kb], arow[kb + 1] };
    v2f b = { brow[kb], brow[kb + 1] };
    c = wmma_f32(a, b, c);
  }
  #pragma unroll
  for (int r = 0; r < 8; ++r) {
    int m = r + hs * 8;
    M1[(tr * 16 + m) * DM + tc * 16 + mn] = c[r];
  }
}

// -------- K6: SE token means (M1 and skip*xn chunk) per (br,b) --------
__global__ void k_se_sums(const float* __restrict__ M1, const float* __restrict__ xn,
                          const float* __restrict__ skip, float* __restrict__ sums) {
  int flat = blockIdx.x;                // br*8+b
  int br = flat >> 3, b = flat & 7;
  int j = threadIdx.x & 63, seg = threadIdx.x >> 6;
  float p = 0.0f;
  if (j < DM) {
    int base = flat * L_SZ;
    for (int l = seg; l < L_SZ; l += 4) p += M1[(base + l) * DM + j];
  } else {
    int jj = j - DM;
    for (int l = seg; l < L_SZ; l += 4) p += xn[(b * L_SZ + l) * C_SZ + br * DM + jj];
  }
  __shared__ float red[256];
  red[threadIdx.x] = p; __syncthreads();
  if (seg == 0) {
    float t = red[j] + red[64 + j] + red[128 + j] + red[192 + j];
    t *= (1.0f / L_SZ);
    if (j >= DM) t *= skip[0];          // M2 = skip_scale * xi
    sums[flat * 64 + j] = t;
  }
}

// -------- K7: SE gates: sigmoid(relu(m @ W1^T) @ W2^T) --------
__global__ void k_se_gates(const float* __restrict__ sums, const float* __restrict__ W1,
                           const float* __restrict__ W2, float* __restrict__ gates) {
  int flat = blockIdx.x;
  int j = threadIdx.x;                  // 64
  __shared__ float m[64];
  m[j] = sums[flat * 64 + j];
  __syncthreads();
  float h0 = 0.0f, h1 = 0.0f;
  #pragma unroll 8
  for (int q = 0; q < 64; ++q) { h0 += m[q] * W1[q]; h1 += m[q] * W1[64 + q]; }
  h0 = fmaxf(h0, 0.0f); h1 = fmaxf(h1, 0.0f);
  float y = h0 * W2[j * 2 + 0] + h1 * W2[j * 2 + 1];
  gates[flat * 64 + j] = sigmoidf_(y);
}

// -------- K8: gated combine (concat) + LayerNorm --------
__global__ void k_gate_ln(const float* __restrict__ M1, const float* __restrict__ xn,
                          const float* __restrict__ gates, const float* __restrict__ skip,
                          const float* __restrict__ g, const float* __restrict__ bt,
                          float* __restrict__ xmn) {
  int bl = blockIdx.x;                  // b*L + l
  int b = bl >> 12, l = bl & (L_SZ - 1);
  int c = threadIdx.x;                  // 128
  int br = c >> 5, j = c & 31;
  int flat = br * 8 + b;
  float g1 = gates[flat * 64 + j];
  float g2 = gates[flat * 64 + 32 + j];
  float m1 = M1[(flat * L_SZ + l) * DM + j];
  float m2 = skip[0] * xn[bl * C_SZ + c];
  float v = g1 * m1 + g2 * m2;
  __shared__ float red[C_SZ];
  red[c] = v; __syncthreads();
  for (int o = 64; o > 0; o >>= 1) { if (c < o) red[c] += red[c + o]; __syncthreads(); }
  float mean = red[0] * (1.0f / C_SZ); __syncthreads();
  float d = v - mean;
  red[c] = d * d; __syncthreads();
  for (int o = 64; o > 0; o >>= 1) { if (c < o) red[c] += red[c + o]; __syncthreads(); }
  float var = red[0] * (1.0f / C_SZ);
  xmn[bl * C_SZ + c] = d * rsqrtf(var + 1e-5f) * g[c] + bt[c];
}

// -------- K9: final projection GEMM (fp32 WMMA) + bias, store NCHW --------
__global__ void k_proj(const float* __restrict__ xmn, const float* __restrict__ pW,
                       const float* __restrict__ pB, float* __restrict__ out) {
  __builtin_prefetch(pW, 0, 1);
  int wave = threadIdx.x >> 5, lane = threadIdx.x & 31;
  int gt = blockIdx.x * 8 + wave;       // 2048 row-tiles * 8 col-tiles
  int tr = gt >> 3, tc = gt & 7;
  int hs = lane >> 4, mn = lane & 15;
  const float* arow = xmn + (tr * 16 + mn) * C_SZ;
  const float* brow = pW + (tc * 16 + mn) * C_SZ;
  v8f c = {};
  #pragma unroll
  for (int kk = 0; kk < 32; ++kk) {     // K = 128 = 32 x 4
    int kb = kk * 4 + hs * 2;
    v2f a = { arow[kb], arow[kb + 1] };
    v2f b = { brow[kb], brow[kb + 1] };
    c = wmma_f32(a, b, c);
  }
  #pragma unroll
  for (int r = 0; r < 8; ++r) {
    int m = r + hs * 8;
    int row = tr * 16 + m;              // b*L + l
    int col = tc * 16 + mn;
    int b = row >> 12, l = row & (L_SZ - 1);
    out[(b * OUT_SZ + col) * L_SZ + l] = c[r] + pB[col];
  }
}

// ---------------- launcher ----------------
extern "C" void kernel_launch(void* const* d_in, const int* in_sizes, int n_in,
                              void* d_out, int out_size, void* d_ws, size_t ws_size,
                              hipStream_t stream) {
  (void)in_sizes; (void)n_in; (void)out_size; (void)ws_size;
  const float* x      = (const float*)d_in[0];
  const float* norm_g = (const float*)d_in[1];
  const float* norm_b = (const float*)d_in[2];
  const float* proj_W = (const float*)d_in[3];
  const float* proj_b = (const float*)d_in[4];
  const float* skip   = (const float*)d_in[5];
  const float* se_W1  = (const float*)d_in[6];
  const float* se_W2  = (const float*)d_in[7];
  const float* in_W   = (const float*)d_in[8];
  const float* conv_W = (const float*)d_in[9];
  const float* conv_b = (const float*)d_in[10];
  const float* xproj_W= (const float*)d_in[11];
  const float* dt_W   = (const float*)d_in[12];
  const float* dt_b   = (const float*)d_in[13];
  const float* A_log  = (const float*)d_in[14];
  const float* D_p    = (const float*)d_in[15];
  const float* out_W  = (const float*)d_in[16];
  float* out = (float*)d_out;
  float* ws  = (float*)d_ws;

  // workspace layout (floats); xraw region reused as ys then xmn;
  // M1 region reused for scan chunk state (dead during scan)
  const size_t BL   = (size_t)B_SZ * L_SZ;        // 32768
  float* xn   = ws;                               // BL*128
  float* xraw = xn   + BL * C_SZ;                 // NB*BL*64
  float* xs   = xraw + (size_t)NB * BL * DI;      // NB*BL*64
  float* zb   = xs   + (size_t)NB * BL * DI;      // NB*BL*64
  float* dtb  = zb   + (size_t)NB * BL * DI;      // NB*BL*64
  float* Bmb  = dtb  + (size_t)NB * BL * DI;      // NB*BL*16
  float* Cmb  = Bmb  + (size_t)NB * BL * DS;      // NB*BL*16
  float* M1   = Cmb  + (size_t)NB * BL * DS;      // NB*BL*32
  float* sums = M1   + (size_t)NB * BL * DM;      // NB*B*64
  float* gates= sums + (size_t)NB * B_SZ * 64;    // NB*B*64
  float* ys   = xraw;                             // reuse (xraw consumed by K3)
  float* xmn  = xraw;                             // reuse (ys consumed by K5)
  const size_t CST = (size_t)NB * B_SZ * NCH * 1024; // 524288 per array
  float* Aprd  = M1;                              // scan chunk state overlaps M1
  float* Hend  = M1 + CST;
  float* Hinit = M1 + 2 * CST;

  k_ln1      <<<BL,              128, 0, stream>>>(x, norm_g, norm_b, xn);
  k_inproj   <<<8192,            256, 0, stream>>>(xn, in_W, xraw, zb);
  k_conv_proj<<<NB * BL,          64, 0, stream>>>(xraw, conv_W, conv_b, xproj_W, dt_W, dt_b,
                                                   xs, dtb, Bmb, Cmb);
  k_scan1    <<<NB * B_SZ * NCH, 1024, 0, stream>>>(dtb, xs, Bmb, A_log, Aprd, Hend);
  k_scan2    <<<NB * B_SZ,       1024, 0, stream>>>(Aprd, Hend, Hinit);
  k_scan3    <<<NB * B_SZ * NCH, 1024, 0, stream>>>(xs, zb, dtb, Bmb, Cmb, A_log, D_p,
                                                    Hinit, ys);
  k_outproj  <<<2048,            256, 0, stream>>>(ys, out_W, M1);
  k_se_sums  <<<NB * B_SZ,       256, 0, stream>>>(M1, xn, skip, sums);
  k_se_gates <<<NB * B_SZ,        64, 0, stream>>>(sums, se_W1, se_W2, gates);
  k_gate_ln  <<<BL,              128, 0, stream>>>(M1, xn, gates, skip, norm_g, norm_b, xmn);
  k_proj     <<<2048,            256, 0, stream>>>(xmn, proj_W, proj_b, out);
}